// unit_gcn_20306605375512
// MI455X (gfx1250) — compile-verified
//
#include <hip/hip_runtime.h>
#include <math.h>

// Problem constants (from reference)
#define C_IN   64
#define T_DIM  256
#define V_DIM  25
#define C_OUT  256
#define S_DIM  3
#define K_DIM  (S_DIM * C_IN)      // 192 = GEMM K (s,c fused)
#define TT     4                   // t-values per workgroup tile
#define NJ     (TT * V_DIM)        // 100 valid (t,w) positions per tile
#define NJP    112                 // padded to 7 * 16
#define NTILES 7                   // WMMA N-tiles per M-tile
#define BN_EPS 1e-5f

typedef float v2f __attribute__((ext_vector_type(2)));
typedef float v8f __attribute__((ext_vector_type(8)));

// ---------------------------------------------------------------------------
// Kernel 1: A_personalized = tanh(A_pers[:n]) -> d_out (output #1),
//           A_final = A + A_personalized -> ws, and zero the stat buffers.
// ---------------------------------------------------------------------------
__global__ void prep_kernel(const float* __restrict__ A,
                            const float* __restrict__ A_pers,
                            float* __restrict__ out_apers,
                            float* __restrict__ Af,
                            float* __restrict__ S1,
                            float* __restrict__ S2,
                            int total) {
    int idx = blockIdx.x * blockDim.x + threadIdx.x;
    if (blockIdx.x == 0 && threadIdx.x < C_OUT) {
        S1[threadIdx.x] = 0.0f;
        S2[threadIdx.x] = 0.0f;
    }
    if (idx < total) {
        float ap = tanhf(A_pers[idx]);
        out_apers[idx] = ap;
        // A has exactly S*V*V = 1875 elements; per-sample offset is idx % 1875
        Af[idx] = A[idx % (S_DIM * V_DIM * V_DIM)] + ap;
    }
}

// ---------------------------------------------------------------------------
// Kernel 2 (fused): per (n, t-tile of 4):
//   phase 1: stage x tile (64c x 4t x 25v) and A_final (transposed) into LDS
//   phase 2: xa[k=(s,c)][j=(t,w)] = sum_v x * Af  -> LDS (192 x 112, padded 0)
//   phase 3: y_raw[o][j] = sum_k conv_w[o][k] * xa[k][j] via f32 WMMA 16x16x4
//            + per-channel sum / sum-of-squares partial reductions
// ---------------------------------------------------------------------------
__launch_bounds__(256, 1)
__global__ void gemm_kernel(const float* __restrict__ x,
                            const float* __restrict__ Af,
                            const float* __restrict__ conv_w,
                            float* __restrict__ y_raw,
                            float* __restrict__ S1,
                            float* __restrict__ S2) {
    extern __shared__ float smem[];
    float* lds_x  = smem;                                    // 64*100   = 6400
    float* lds_xa = smem + C_IN * NJ;                        // 192*112  = 21504
    float* lds_af = lds_xa + K_DIM * NJP;                    // 3*25*25  = 1875

    const int bx  = blockIdx.x;
    const int n   = bx >> 6;            // T_DIM/TT = 64 t-tiles per sample
    const int tt  = bx & 63;
    const int t0  = tt * TT;
    const int tid = threadIdx.x;

    // ---- phase 1: stage inputs ----
    {
        const float* xsrc = x + (size_t)n * C_IN * T_DIM * V_DIM;
        for (int i = tid; i < C_IN * NJ; i += 256) {
            int c = i / NJ, r = i - c * NJ;   // r = tl*25 + v, contiguous in mem
            lds_x[i] = xsrc[(size_t)c * (T_DIM * V_DIM) + t0 * V_DIM + r];
        }
        const float* asrc = Af + (size_t)n * (S_DIM * V_DIM * V_DIM);
        for (int i = tid; i < S_DIM * V_DIM * V_DIM; i += 256) {
            int s = i / 625, r = i - s * 625;
            int w = r / V_DIM, v = r - w * V_DIM;
            // store transposed: lds_af[s][w][v] so phase-2 reads are contiguous
            lds_af[i] = asrc[s * 625 + v * V_DIM + w];
        }
        // zero pad columns j = 100..111 so WMMA on them accumulates zeros
        for (int i = tid; i < K_DIM * (NJP - NJ); i += 256) {
            int k = i / (NJP - NJ), j = NJ + (i - k * (NJP - NJ));
            lds_xa[k * NJP + j] = 0.0f;
        }
    }
    __syncthreads();

    // ---- phase 2: graph aggregation xa = x @ A_final ----
    for (int o = tid; o < K_DIM * NJ; o += 256) {
        int k = o / NJ, j = o - k * NJ;
        int s = k >> 6, c = k & 63;
        int tl = j / V_DIM, w = j - tl * V_DIM;
        const float* xr = &lds_x[(c * TT + tl) * V_DIM];
        const float* ar = &lds_af[(s * V_DIM + w) * V_DIM];
        float acc = 0.0f;
#pragma unroll
        for (int v = 0; v < V_DIM; ++v) acc = fmaf(xr[v], ar[v], acc);
        lds_xa[k * NJP + j] = acc;
    }
    __syncthreads();

    // ---- phase 3: WMMA GEMM: M=256 (C_OUT) x N=112 x K=192 ----
    const int wv   = tid >> 5;     // wave id 0..7
    const int lane = tid & 31;
    const int half = lane >> 4;    // 0: lanes 0-15, 1: lanes 16-31
    const int lm   = lane & 15;

    for (int mi = 0; mi < 2; ++mi) {
        const int mt   = wv + mi * 8;      // M-tile 0..15
        const int orow = mt * 16 + lm;     // A-matrix row (output channel)
        v8f acc[NTILES] = {};              // zero accumulators

        for (int k0 = 0; k0 < K_DIM; k0 += 4) {
            // A fragment (conv_w): lane holds row orow, K = k0 + {0,1}/{2,3}
            const int s  = k0 >> 6;
            const int cb = (k0 & 63) + half * 2;   // even -> 8B aligned
            v2f a = *reinterpret_cast<const v2f*>(
                &conv_w[((size_t)s * C_OUT + orow) * C_IN + cb]);
            const int kr = k0 + half * 2;
#pragma unroll
            for (int nt = 0; nt < NTILES; ++nt) {
                // B fragment (xa): lane holds col nt*16+lm, rows kr, kr+1
                v2f b;
                b.x = lds_xa[kr * NJP + nt * 16 + lm];
                b.y = lds_xa[(kr + 1) * NJP + nt * 16 + lm];
                acc[nt] = __builtin_amdgcn_wmma_f32_16x16x4_f32(
                    false, a, false, b, (short)0, acc[nt], false, false);
            }
        }

        // ---- store y_raw (valid j only) ----
#pragma unroll
        for (int nt = 0; nt < NTILES; ++nt) {
            int j = nt * 16 + lm;
            if (j < NJ) {
#pragma unroll
                for (int r = 0; r < 8; ++r) {
                    int o = mt * 16 + r + half * 8;   // C layout: M = r + 8*half
                    y_raw[(size_t)(n * C_OUT + o) * (T_DIM * V_DIM)
                          + t0 * V_DIM + j] = acc[nt][r];
                }
            }
        }

        // ---- BN statistics: per-channel sum & sum-of-squares ----
        // (padded columns hold exact zeros -> contribute nothing)
#pragma unroll
        for (int r = 0; r < 8; ++r) {
            float p1 = 0.0f, p2 = 0.0f;
#pragma unroll
            for (int nt = 0; nt < NTILES; ++nt) {
                float v = acc[nt][r];
                p1 += v;
                p2 += v * v;
            }
            // reduce across the 16 lanes of each half (halves stay disjoint)
#pragma unroll
            for (int m = 1; m < 16; m <<= 1) {
                p1 += __shfl_xor(p1, m, 32);
                p2 += __shfl_xor(p2, m, 32);
            }
            if (lm == 0) {
                int o = mt * 16 + r + half * 8;
                atomicAdd(&S1[o], p1);
                atomicAdd(&S2[o], p2);
            }
        }
    }
}

// ---------------------------------------------------------------------------
// Kernel 3: fold BN stats into per-channel affine (conv bias cancels: the
// batch mean absorbs it, so (y+b-mean)*scale == (y - S1/cnt)*scale).
// ---------------------------------------------------------------------------
__global__ void stats_kernel(const float* __restrict__ S1,
                             const float* __restrict__ S2,
                             const float* __restrict__ bn_w,
                             const float* __restrict__ bn_b,
                             float* __restrict__ alpha,
                             float* __restrict__ beta,
                             float inv_cnt) {
    int o = threadIdx.x;
    float mu  = S1[o] * inv_cnt;
    float var = S2[o] * inv_cnt - mu * mu;
    float sc  = bn_w[o] * rsqrtf(var + BN_EPS);
    alpha[o] = sc;
    beta[o]  = bn_b[o] - mu * sc;
}

// ---------------------------------------------------------------------------
// Kernel 4: y = relu(y_raw * alpha[o] + beta[o]), in place, float4 vectorized.
// ---------------------------------------------------------------------------
__global__ void finalize_kernel(float* __restrict__ y,
                                const float* __restrict__ alpha,
                                const float* __restrict__ beta,
                                size_t total4) {
    size_t i = (size_t)blockIdx.x * blockDim.x + threadIdx.x;
    if (i >= total4) return;
    float4* p = reinterpret_cast<float4*>(y);
    float4 v = p[i];
    int o = (int)((i * 4 / (T_DIM * V_DIM)) % C_OUT);
    float a = alpha[o], b = beta[o];
    v.x = fmaxf(fmaf(v.x, a, b), 0.0f);
    v.y = fmaxf(fmaf(v.y, a, b), 0.0f);
    v.z = fmaxf(fmaf(v.z, a, b), 0.0f);
    v.w = fmaxf(fmaf(v.w, a, b), 0.0f);
    p[i] = v;
}

// ---------------------------------------------------------------------------
extern "C" void kernel_launch(void* const* d_in, const int* in_sizes, int n_in,
                              void* d_out, int out_size, void* d_ws, size_t ws_size,
                              hipStream_t stream) {
    const float* x      = (const float*)d_in[0];
    const float* A      = (const float*)d_in[1];
    const float* A_pers = (const float*)d_in[2];
    const float* conv_w = (const float*)d_in[3];
    /* conv_b = d_in[4] is mathematically cancelled by batch-norm */
    const float* bn_w   = (const float*)d_in[5];
    const float* bn_b   = (const float*)d_in[6];

    const int n = in_sizes[0] / (C_IN * T_DIM * V_DIM);   // batch (64)

    float* out = (float*)d_out;
    float* ws  = (float*)d_ws;
    float* Af    = ws;               // n * 1875 floats
    float* S1    = ws + 120000;      // 256
    float* S2    = ws + 120256;      // 256
    float* alpha = ws + 120512;      // 256
    float* beta  = ws + 120768;      // 256

    const int apers_total = n * S_DIM * V_DIM * V_DIM;    // output #1 size
    float* y_raw = out + apers_total;                      // output #2 region

    prep_kernel<<<(apers_total + 255) / 256, 256, 0, stream>>>(
        A, A_pers, out, Af, S1, S2, apers_total);

    const int ttiles = T_DIM / TT;                        // 64
    const size_t lds_bytes =
        (size_t)(C_IN * NJ + K_DIM * NJP + S_DIM * V_DIM * V_DIM) * sizeof(float);
    gemm_kernel<<<n * ttiles, 256, lds_bytes, stream>>>(
        x, Af, conv_w, y_raw, S1, S2);

    float inv_cnt = 1.0f / (float)(n * T_DIM * V_DIM);
    stats_kernel<<<1, C_OUT, 0, stream>>>(S1, S2, bn_w, bn_b, alpha, beta, inv_cnt);

    size_t total4 = ((size_t)n * C_OUT * T_DIM * V_DIM) / 4;
    finalize_kernel<<<(unsigned)((total4 + 255) / 256), 256, 0, stream>>>(
        y_raw, alpha, beta, total4);
}